// GCN_44495861186899
// MI455X (gfx1250) — compile-verified
//
#include <hip/hip_runtime.h>
#include <cmath>

typedef __attribute__((ext_vector_type(2))) float v2f;
typedef __attribute__((ext_vector_type(8))) float v8f;

// ---------------------------------------------------------------------------
// fp32 WMMA GEMM: C[M x NC] = A[M x K] @ B[K x NC], row-major, one wave per
// 16x16 output tile, K multiple of 4.  Uses V_WMMA_F32_16X16X4_F32.
// A layout : lane m=lane&15 holds row m, K = 2*(lane>>4) + {0,1}
// B layout : lane n=lane&15 holds col n, K = 2*(lane>>4) + {0,1}
// C layout : lane n=lane&15 holds col n, M = 8*(lane>>4) + v (v = 0..7)
// ---------------------------------------------------------------------------
__global__ __launch_bounds__(32) void gcn_wmma_gemm_f32(
    const float* __restrict__ A, const float* __restrict__ B,
    float* __restrict__ C, int K, int NC)
{
    const int lane = threadIdx.x;
    const int r    = lane & 15;
    const int kh   = lane >> 4;
    const int rowA = blockIdx.x * 16 + r;
    const int col  = blockIdx.y * 16 + r;

    v8f acc = {};
    for (int k0 = 0; k0 < K; k0 += 4) {
        const int ka = k0 + 2 * kh;
        v2f a, b;
        a[0] = A[rowA * K + ka];
        a[1] = A[rowA * K + ka + 1];
        b[0] = B[ka * NC + col];
        b[1] = B[(ka + 1) * NC + col];
        acc = __builtin_amdgcn_wmma_f32_16x16x4_f32(
            false, a, false, b, (short)0, acc, false, false);
    }
#pragma unroll
    for (int v = 0; v < 8; ++v) {
        const int row = blockIdx.x * 16 + 8 * kh + v;
        C[row * NC + col] = acc[v];
    }
}

// ---------------------------------------------------------------------------
// Per-timestep init: deg = 1.0 (self-loop weight), hagg = 0, zagg = 0
// ---------------------------------------------------------------------------
__global__ void gcn_init_kernel(float* __restrict__ deg,
                                float* __restrict__ hagg,
                                float* __restrict__ zagg, int N)
{
    const int i = blockIdx.x * blockDim.x + threadIdx.x;
    if (i < N) {
        deg[i] = 1.0f;
    } else if (i < N + N * 64) {
        hagg[i - N] = 0.0f;
    } else if (i < N + N * 64 + N * 32) {
        zagg[i - N - N * 64] = 0.0f;
    }
}

// deg[dst] += ew  (self-loop "1" already in init)
__global__ void gcn_deg_kernel(const int* __restrict__ dst,
                               const float* __restrict__ ew,
                               float* __restrict__ deg, int E)
{
    const int e = blockIdx.x * blockDim.x + threadIdx.x;
    if (e < E) atomicAdd(&deg[dst[e]], ew[e]);
}

__global__ void gcn_dinv_kernel(const float* __restrict__ deg,
                                float* __restrict__ dinv, int N)
{
    const int i = blockIdx.x * blockDim.x + threadIdx.x;
    if (i < N) {
        const float d = deg[i];
        dinv[i] = (d > 0.0f) ? rsqrtf(d) : 0.0f;
    }
}

// ---------------------------------------------------------------------------
// Edge scatter: out[dst][k] += in[src][k] * dinv[src]*w*dinv[dst]
// One thread per (edge, dim); src/dst/w are wave-uniform -> scalar broadcast,
// atomics are coalesced across dims.  D = 1<<shift dims per row.
// ---------------------------------------------------------------------------
__global__ void gcn_scatter_kernel(const float* __restrict__ in,
                                   const int* __restrict__ src,
                                   const int* __restrict__ dst,
                                   const float* __restrict__ ew,
                                   const float* __restrict__ dinv,
                                   float* __restrict__ out,
                                   int E, int shift)
{
    const long long gid = (long long)blockIdx.x * blockDim.x + threadIdx.x;
    const int e = (int)(gid >> shift);
    const int k = (int)(gid & ((1 << shift) - 1));
    if (e < E) {
        const int s = src[e];
        const int d = dst[e];
        const float nrm = dinv[s] * ew[e] * dinv[d];
        atomicAdd(&out[((long long)d << shift) + k],
                  in[((long long)s << shift) + k] * nrm);
    }
}

// h = relu(hagg + hx*dinv^2 + b1), in place in hagg (64 dims)
__global__ void gcn_epi1_kernel(float* __restrict__ hagg,
                                const float* __restrict__ hx,
                                const float* __restrict__ dinv,
                                const float* __restrict__ b1, int N)
{
    const int i = blockIdx.x * blockDim.x + threadIdx.x;
    if (i < N * 64) {
        const int n = i >> 6;
        const float di = dinv[n];
        const float v = hagg[i] + hx[i] * di * di + b1[i & 63];
        hagg[i] = fmaxf(v, 0.0f);
    }
}

// out[t] = tanh(zagg + hz*dinv^2 + b2)  (32 dims)
__global__ void gcn_epi2_kernel(const float* __restrict__ zagg,
                                const float* __restrict__ hz,
                                const float* __restrict__ dinv,
                                const float* __restrict__ b2,
                                float* __restrict__ out, int N)
{
    const int i = blockIdx.x * blockDim.x + threadIdx.x;
    if (i < N * 32) {
        const int n = i >> 5;
        const float di = dinv[n];
        const float v = zagg[i] + hz[i] * di * di + b2[i & 31];
        out[i] = tanhf(v);
    }
}

static inline int cdiv(long long a, int b) { return (int)((a + b - 1) / b); }

extern "C" void kernel_launch(void* const* d_in, const int* in_sizes, int n_in,
                              void* d_out, int out_size, void* d_ws, size_t ws_size,
                              hipStream_t stream)
{
    const float* x  = (const float*)d_in[0];   // [N,128]
    const int*   ei = (const int*)d_in[1];     // [T,2,E]
    const float* ew = (const float*)d_in[2];   // [T,E]
    const float* W1 = (const float*)d_in[3];   // [128,64]
    const float* b1 = (const float*)d_in[4];   // [64]
    const float* W2 = (const float*)d_in[5];   // [64,32]
    const float* b2 = (const float*)d_in[6];   // [32]
    float* out = (float*)d_out;                // [T,N,32]

    const int N = in_sizes[0] / 128;
    const int T = out_size / (N * 32);
    const int E = in_sizes[2] / T;

    float* ws   = (float*)d_ws;
    float* hx   = ws;              ws += (size_t)N * 64;  // x @ W1
    float* deg  = ws;              ws += N;
    float* dinv = ws;              ws += N;
    float* hagg = ws;              ws += (size_t)N * 64;  // layer-1 agg, then h
    float* hz   = ws;              ws += (size_t)N * 32;  // h @ W2
    float* zagg = ws;              ws += (size_t)N * 32;  // layer-2 agg
    (void)ws_size; (void)n_in;

    // hx = x @ W1  (edge-independent: hoisted out of the t-loop)
    gcn_wmma_gemm_f32<<<dim3(N / 16, 64 / 16), 32, 0, stream>>>(x, W1, hx, 128, 64);

    for (int t = 0; t < T; ++t) {
        const int*   src_t = ei + (size_t)t * 2 * E;
        const int*   dst_t = src_t + E;
        const float* ew_t  = ew + (size_t)t * E;

        gcn_init_kernel<<<cdiv((long long)N * 97, 256), 256, 0, stream>>>(deg, hagg, zagg, N);
        gcn_deg_kernel<<<cdiv(E, 256), 256, 0, stream>>>(dst_t, ew_t, deg, E);
        gcn_dinv_kernel<<<cdiv(N, 256), 256, 0, stream>>>(deg, dinv, N);

        // layer 1: scatter hx, then bias+relu (self-loop folded into epilogue)
        gcn_scatter_kernel<<<cdiv((long long)E * 64, 256), 256, 0, stream>>>(
            hx, src_t, dst_t, ew_t, dinv, hagg, E, 6);
        gcn_epi1_kernel<<<cdiv((long long)N * 64, 256), 256, 0, stream>>>(hagg, hx, dinv, b1, N);

        // layer 2: hz = h @ W2, scatter, bias+tanh -> out[t]
        gcn_wmma_gemm_f32<<<dim3(N / 16, 32 / 16), 32, 0, stream>>>(hagg, W2, hz, 64, 32);
        gcn_scatter_kernel<<<cdiv((long long)E * 32, 256), 256, 0, stream>>>(
            hz, src_t, dst_t, ew_t, dinv, zagg, E, 5);
        gcn_epi2_kernel<<<cdiv((long long)N * 32, 256), 256, 0, stream>>>(
            zagg, hz, dinv, b2, out + (size_t)t * N * 32, N);
    }
}